// Custom_BMN_22900765622418
// MI455X (gfx1250) — compile-verified
//
#include <hip/hip_runtime.h>
#include <hip/hip_bf16.h>

// ---------------- problem constants ----------------
#define SEQ   320          // T / t_start dim
#define FEAT  2048
#define HID   512
#define NB    4
#define NC    21           // NCLASS+1
#define DSC   50           // TSCALE
#define MASKC 512000       // NSAMP*DSC*SEQ columns of sample_mask
#define GCOLS 48           // 3 groups * 16 s' per block tile

#define OFF_YATN  (NB*NC*SEQ)                 // 26880
#define OFF_BMN   (OFF_YATN + NB*SEQ)         // 28160

typedef float v2f __attribute__((ext_vector_type(2)));
typedef float v8f __attribute__((ext_vector_type(8)));

__device__ __forceinline__ int iclamp(int v, int lo, int hi) {
    return v < lo ? lo : (v > hi ? hi : v);
}

// Native fp32 WMMA: D(16x16,f32) = A(16x4,f32) * B(4x16,f32) + C
__device__ __forceinline__ v8f wmma_f32x4(v2f a, v2f b, v8f c) {
    return __builtin_amdgcn_wmma_f32_16x16x4_f32(
        /*neg_a=*/false, a, /*neg_b=*/false, b,
        /*c_mod=*/(short)0, c, /*reuse_a=*/false, /*reuse_b=*/false);
}

// ---------------------------------------------------------------------------
// Kernel 0: repack w1 [HID,FEAT,3] -> wp [3][HID][FEAT] (contiguous in cin)
// so WMMA A-fragments load as a single b64.
// ---------------------------------------------------------------------------
__global__ void repack_w1(const float* __restrict__ w1, float* __restrict__ wp) {
    const int idx = blockIdx.x * blockDim.x + threadIdx.x;   // 3*512*2048
    if (idx >= 3 * HID * FEAT) return;
    const int cin = idx % FEAT;
    const int h   = (idx / FEAT) % HID;
    const int kw  = idx / (FEAT * HID);
    wp[idx] = w1[((size_t)h * FEAT + cin) * 3 + kw];
}

// ---------------------------------------------------------------------------
// Kernel 1: feat[b,h,t] = relu(b1[h] + sum_{cin,kw} x[b,t+kw-1,cin]*wp[kw,h,cin])
// One wave == one 32(h) x 32(t) tile: 2x2 fragment blocking, 4 WMMAs per
// 4 b64 loads. Boundary handled branchlessly (clamped row + 0/1 mask mult).
// jobs = B(4) * HT(16) * TT(10) = 640 waves = 80 blocks * 8 waves.
// ---------------------------------------------------------------------------
__global__ __launch_bounds__(256) void conv1_relu_wmma(
    const float* __restrict__ x, const float* __restrict__ wp,
    const float* __restrict__ b1, float* __restrict__ feat) {
    const int wave = threadIdx.x >> 5;
    const int lane = threadIdx.x & 31;
    const int job  = blockIdx.x * 8 + wave;
    const int tt = job % 10;
    const int ht = (job / 10) % 16;
    const int b  = job / 160;

    const int lm = lane & 15;          // A: row  |  B: col (within 16-tile)
    const int kb = (lane >> 4) * 2;    // K-pair select (ISA 16x4 / 4x16 layout)
    const int h0 = ht * 32 + lm, h1 = h0 + 16;
    const int t0 = tt * 32 + lm, t1 = t0 + 16;

    v8f acc00 = {}, acc01 = {}, acc10 = {}, acc11 = {};
    for (int kw = 0; kw < 3; ++kw) {
        const int tp0 = t0 + kw - 1, tp1 = t1 + kw - 1;
        const float m0 = (tp0 >= 0 && tp0 < SEQ) ? 1.f : 0.f;
        const float m1 = (tp1 >= 0 && tp1 < SEQ) ? 1.f : 0.f;
        const float* xr0 = x + (size_t)(b * SEQ + iclamp(tp0, 0, SEQ - 1)) * FEAT + kb;
        const float* xr1 = x + (size_t)(b * SEQ + iclamp(tp1, 0, SEQ - 1)) * FEAT + kb;
        const float* wr0 = wp + (size_t)(kw * HID + h0) * FEAT + kb;
        const float* wr1 = wp + (size_t)(kw * HID + h1) * FEAT + kb;
        #pragma unroll 2
        for (int c0 = 0; c0 < FEAT; c0 += 4) {
            v2f a0 = *(const v2f*)(wr0 + c0);          // global_load_b64
            v2f a1 = *(const v2f*)(wr1 + c0);
            v2f bv0 = *(const v2f*)(xr0 + c0);
            v2f bv1 = *(const v2f*)(xr1 + c0);
            bv0.x *= m0; bv0.y *= m0;                  // branchless edge zeroing
            bv1.x *= m1; bv1.y *= m1;
            acc00 = wmma_f32x4(a0, bv0, acc00);
            acc01 = wmma_f32x4(a0, bv1, acc01);
            acc10 = wmma_f32x4(a1, bv0, acc10);
            acc11 = wmma_f32x4(a1, bv1, acc11);
        }
    }
    // bias + relu + store (C/D layout: lane holds col lm, rows v + (lane>=16)*8)
    const int r0 = ht * 32 + (lane >> 4) * 8;          // rows of acc0x
    const int r1 = r0 + 16;                            // rows of acc1x
    #pragma unroll
    for (int v = 0; v < 8; ++v) {
        const float bia0 = b1[r0 + v], bia1 = b1[r1 + v];
        float v00 = acc00[v] + bia0; v00 = v00 > 0.f ? v00 : 0.f;
        float v01 = acc01[v] + bia0; v01 = v01 > 0.f ? v01 : 0.f;
        float v10 = acc10[v] + bia1; v10 = v10 > 0.f ? v10 : 0.f;
        float v11 = acc11[v] + bia1; v11 = v11 > 0.f ? v11 : 0.f;
        feat[(size_t)(b * HID + r0 + v) * SEQ + t0] = v00;
        feat[(size_t)(b * HID + r0 + v) * SEQ + t1] = v01;
        feat[(size_t)(b * HID + r1 + v) * SEQ + t0] = v10;
        feat[(size_t)(b * HID + r1 + v) * SEQ + t1] = v11;
    }
}

// ---------------------------------------------------------------------------
// Kernel 2: y_class[b,c,t] = sum_h wc[c,h] * feat[b,h,t]   (no bias)
// ---------------------------------------------------------------------------
__global__ void yclass_kernel(const float* __restrict__ feat,
                              const float* __restrict__ wc,
                              float* __restrict__ yc_ws,
                              float* __restrict__ out) {
    const int idx = blockIdx.x * blockDim.x + threadIdx.x;   // 26880 threads
    if (idx >= NB * NC * SEQ) return;
    const int t = idx % SEQ;
    const int c = (idx / SEQ) % NC;
    const int b = idx / (SEQ * NC);
    const float* f = feat + (size_t)b * HID * SEQ + t;
    const float* w = wc + c * HID;
    float sum = 0.f;
    #pragma unroll 8
    for (int h = 0; h < HID; ++h) sum += f[(size_t)h * SEQ] * w[h];
    out[idx]   = sum;
    yc_ws[idx] = sum;
}

// ---------------------------------------------------------------------------
// Kernel 3: y_atn[b,t] = ba + sum_h sum_kw feat[b,h,t+kw-1]*wa[h,kw]  (pad=1)
// ---------------------------------------------------------------------------
__global__ void yatn_kernel(const float* __restrict__ feat,
                            const float* __restrict__ wa,
                            const float* __restrict__ ba,
                            float* __restrict__ out) {
    const int idx = blockIdx.x * blockDim.x + threadIdx.x;   // 1280 threads
    if (idx >= NB * SEQ) return;
    const int t = idx % SEQ;
    const int b = idx / SEQ;
    const float* f = feat + (size_t)b * HID * SEQ;
    const float mL = (t > 0) ? 1.f : 0.f;
    const float mR = (t < SEQ - 1) ? 1.f : 0.f;
    const int tL = iclamp(t - 1, 0, SEQ - 1), tR = iclamp(t + 1, 0, SEQ - 1);
    float sum = ba[0];
    for (int h = 0; h < HID; ++h) {
        sum += f[(size_t)h * SEQ + tL] * wa[h * 3]     * mL;
        sum += f[(size_t)h * SEQ + t ] * wa[h * 3 + 1];
        sum += f[(size_t)h * SEQ + tR] * wa[h * 3 + 2] * mR;
    }
    out[OFF_YATN + idx] = sum;
}

// ---------------------------------------------------------------------------
// Kernel 4 (fused boundary matching): one block per (d, s'-tile of 16).
// Phase 1: stream mask[s, k, d, s'] exactly once, reduce over k into 3 group
//          means -> LDS tile red[320 s][3 g * 16 s']  (61.4 KB).
// Phase 2: WMMA GEMM  bmn[b,g,c,d,s'] = sum_s y_class[b,c,s] * red[s][g,s'].
//          M = B*C = 84 (padded to 96 -> 6 m-tiles) x 3 g = 18 wave-jobs.
// ---------------------------------------------------------------------------
__global__ __launch_bounds__(256) void bm_fused(
    const float* __restrict__ mask, const float* __restrict__ yclass,
    float* __restrict__ out) {
    __shared__ float red[SEQ * GCOLS];                 // 61440 bytes
    const int d  = blockIdx.x / 20;
    const int st = blockIdx.x % 20;
    const int tid = threadIdx.x;

    // ---- phase 1: k-reduction into LDS ----
    {
        const int sp = tid & 15, srow = tid >> 4;      // 16 s' x 16 s per pass
        for (int p = 0; p < 20; ++p) {
            const int s = p * 16 + srow;
            const size_t base = (size_t)s * MASKC + (size_t)d * SEQ + st * 16 + sp;
            if (p + 1 < 20)                            // global_prefetch_b8
                __builtin_prefetch(&mask[base + (size_t)16 * MASKC], 0, 1);
            float s0 = 0.f, s1 = 0.f, s2 = 0.f;
            #pragma unroll
            for (int k = 0; k < 8; ++k)   s0 += mask[base + (size_t)k * (DSC * SEQ)];
            #pragma unroll
            for (int k = 8; k < 24; ++k)  s1 += mask[base + (size_t)k * (DSC * SEQ)];
            #pragma unroll
            for (int k = 24; k < 32; ++k) s2 += mask[base + (size_t)k * (DSC * SEQ)];
            red[s * GCOLS + sp]      = s0 * 0.125f;
            red[s * GCOLS + 16 + sp] = s1 * 0.0625f;
            red[s * GCOLS + 32 + sp] = s2 * 0.125f;
        }
    }
    __syncthreads();

    // ---- phase 2: WMMA GEMM out of LDS ----
    const int wave = tid >> 5, lane = tid & 31;
    const int lm = lane & 15;
    const int kb = (lane >> 4) * 2;
    for (int job = wave; job < 18; job += 8) {
        const int mt = job / 3, g = job % 3;
        const int m = mt * 16 + lm;                    // row of y_class (b*21+c)
        const float mokf = (m < NB * NC) ? 1.f : 0.f;  // branchless M-pad guard
        const float* arow = yclass + (size_t)iclamp(m, 0, NB * NC - 1) * SEQ + kb;
        v8f acc = {};
        for (int k0 = 0; k0 < SEQ; k0 += 4) {
            v2f a = *(const v2f*)(arow + k0);          // global_load_b64
            a.x *= mokf; a.y *= mokf;
            v2f bb;
            bb.x = red[(k0 + kb) * GCOLS + g * 16 + lm];
            bb.y = red[(k0 + kb + 1) * GCOLS + g * 16 + lm];
            acc = wmma_f32x4(a, bb, acc);
        }
        const int spcol = st * 16 + lm;
        const int mbase = mt * 16 + (lane >> 4) * 8;
        #pragma unroll
        for (int v = 0; v < 8; ++v) {
            const int mm = mbase + v;
            if (mm < NB * NC) {
                const int b = mm / NC, c = mm % NC;
                out[OFF_BMN + ((size_t)((b * 3 + g) * NC + c)) * (DSC * SEQ)
                            + d * SEQ + spcol] = acc[v];
            }
        }
    }
}

// ---------------------------------------------------------------------------
extern "C" void kernel_launch(void* const* d_in, const int* in_sizes, int n_in,
                              void* d_out, int out_size, void* d_ws, size_t ws_size,
                              hipStream_t stream) {
    const float* x    = (const float*)d_in[0];   // [4,320,2048]
    const float* w1   = (const float*)d_in[1];   // [512,2048,3]
    const float* b1   = (const float*)d_in[2];   // [512]
    const float* wc   = (const float*)d_in[3];   // [21,512,1]
    const float* wa   = (const float*)d_in[4];   // [1,512,3]
    const float* ba   = (const float*)d_in[5];   // [1]
    const float* mask = (const float*)d_in[6];   // [320,512000]
    float* out = (float*)d_out;

    float* feat = (float*)d_ws;                       // [4,512,320]   2.62 MB
    float* yc   = feat + (size_t)NB * HID * SEQ;      // [4,21,320]    0.1 MB
    float* wp   = yc + (size_t)NB * NC * SEQ;         // [3,512,2048] 12.6 MB

    repack_w1      <<<(3 * HID * FEAT + 255) / 256, 256, 0, stream>>>(w1, wp);
    conv1_relu_wmma<<<80, 256, 0, stream>>>(x, wp, b1, feat);
    yclass_kernel  <<<(NB * NC * SEQ + 255) / 256, 256, 0, stream>>>(feat, wc, yc, out);
    yatn_kernel    <<<(NB * SEQ + 255) / 256, 256, 0, stream>>>(feat, wa, ba, out);
    bm_fused       <<<DSC * 20, 256, 0, stream>>>(mask, yc, out);
}